// GraphAttention_19713899889134
// MI455X (gfx1250) — compile-verified
//
#include <hip/hip_runtime.h>
#include <hip/hip_bf16.h>

// GraphAttention for MI455X (gfx1250, wave32).
//
// Roofline: output 67MB write + 33.5MB mask read (L2-resident across heads)
// => ~100MB HBM => ~4.3us floor @ 23.3 TB/s. Main kernel is bandwidth-bound;
// ~33M v_exp_f32 hide under memory latency. The f1/f2 projection (the only
// matmul) uses V_WMMA_F32_16X16X4_F32 chained over K=64.

#define GA_H 4
#define GA_N 2048
#define GA_D 64

typedef __attribute__((ext_vector_type(2))) float v2f;
typedef __attribute__((ext_vector_type(8))) float v8f;

// ---------------------------------------------------------------------------
// Kernel 1: f1/f2 projection via WMMA f32 16x16x4.
//   D(16x16) accumulates A(16x4) x B(4x16) over 16 K-chunks (K total = 64).
//   B column 0 = V[h,0,:] chunk, column 1 = V[h,1,:] chunk, columns 2..15 = 0,
//   so D[:,0] = f1 tile and D[:,1] = f2 tile after accumulation.
// A layout (ISA 7.12.2, 32-bit A 16x4): lane m(0..15) holds K=4c+0/4c+1 in
// v0/v1; lane m+16 holds K=4c+2/4c+3. B mirrors this (row-striped per VGPR).
// C/D layout: acc[r] = D[M = 8*half + r, N = lane&15].
// One wave32 block per 16-row tile => EXEC all-ones as WMMA requires.
// F layout in workspace: F[0][h][n] = f1, F[1][h][n] = f2.
// ---------------------------------------------------------------------------
__global__ __launch_bounds__(32) void ga_proj_wmma(
    const float* __restrict__ X,   // [H, N, D]
    const float* __restrict__ V,   // [H, 2, D]
    float* __restrict__ F)         // [2, H, N]
{
    const int tile = blockIdx.x;            // 0 .. H*(N/16)-1
    const int h    = tile / (GA_N / 16);
    const int row0 = (tile % (GA_N / 16)) * 16;
    const int lane = threadIdx.x;           // 0..31 (wave32)
    const int m    = lane & 15;
    const int half = lane >> 4;

    const float* Arow = X + ((size_t)h * GA_N + row0 + m) * GA_D;
    // Safe (in-bounds) V address for every lane; selected below.
    const float* Vrow = V + ((size_t)h * 2 + (m & 1)) * GA_D;

    v8f acc = {};
#pragma unroll
    for (int c = 0; c < 16; ++c) {
        const int k = 4 * c + 2 * half;
        v2f a = *(const v2f*)(Arow + k);
        v2f bz = {0.0f, 0.0f};
        v2f bv = *(const v2f*)(Vrow + k);
        v2f b = (m < 2) ? bv : bz;          // cols >=2 of B are zero
        acc = __builtin_amdgcn_wmma_f32_16x16x4_f32(
            /*neg_a=*/false, a, /*neg_b=*/false, b,
            /*c_mod=*/(short)0, acc, /*reuse_a=*/false, /*reuse_b=*/false);
    }

    // Column m of D: m==0 -> f1, m==1 -> f2. acc[r] = value for row 8*half+r.
    if (m < 2) {
        float* dst = F + (size_t)m * (GA_H * GA_N)
                       + (size_t)h * GA_N + row0 + 8 * half;
#pragma unroll
        for (int r = 0; r < 8; ++r) dst[r] = acc[r];
    }
}

// ---------------------------------------------------------------------------
// Kernel 2: one block per (h, row i). 256 threads, each owns 2 x float4
// column groups (B128 loads/stores, fully coalesced). exp values kept in
// registers between the sum pass and the write pass (no recompute).
// Row softmax without max-shift is exact-safe here: sig in (0,1).
// ---------------------------------------------------------------------------
__global__ __launch_bounds__(256) void ga_att(
    const float* __restrict__ localm,   // [N, N] in {0,1}
    const float* __restrict__ longm,    // [N, N] in {0,1}
    const float* __restrict__ F,        // [2, H, N]
    float* __restrict__ out)            // [H, N, N]
{
    const int b = blockIdx.x;           // 0 .. H*N-1
    const int h = b >> 11;              // / 2048
    const int i = b & (GA_N - 1);
    const int t = threadIdx.x;

    const float f1i = F[(size_t)h * GA_N + i];
    const float* f2row = F + (size_t)(GA_H * GA_N) + (size_t)h * GA_N;
    const float* lrow  = localm + (size_t)i * GA_N;
    const float* grow  = longm  + (size_t)i * GA_N;

    // gfx1250 prefetch hint (global_prefetch_b8); masks are L2-resident.
    __builtin_prefetch(lrow + 4 * t, 0, 0);
    __builtin_prefetch(grow + 4 * t, 0, 0);

    float e[2][4];
    float ml[2][4], mg[2][4];
    float sumL = 0.0f, sumG = 0.0f;

#pragma unroll
    for (int g = 0; g < 2; ++g) {
        const int j = g * 1024 + 4 * t;
        const float4 f2v = *(const float4*)(f2row + j);
        const float4 lv  = *(const float4*)(lrow + j);
        const float4 gv  = *(const float4*)(grow + j);
        const float f2a[4] = {f2v.x, f2v.y, f2v.z, f2v.w};
        const float la[4]  = {lv.x, lv.y, lv.z, lv.w};
        const float ga[4]  = {gv.x, gv.y, gv.z, gv.w};
#pragma unroll
        for (int q = 0; q < 4; ++q) {
            const float s   = f1i + f2a[q];
            const float sig = 1.0f / (1.0f + __expf(-s));   // sigmoid
            const float ee  = __expf(sig);                  // exp(logit), logit=sig
            e[g][q]  = ee;
            ml[g][q] = la[q];
            mg[g][q] = ga[q];
            if (la[q] != 0.0f) sumL += ee;
            if (ga[q] != 0.0f) sumG += ee;
        }
    }

    // wave32 butterfly reduction, then cross-wave via LDS.
#pragma unroll
    for (int off = 16; off > 0; off >>= 1) {
        sumL += __shfl_xor(sumL, off, 32);
        sumG += __shfl_xor(sumG, off, 32);
    }
    __shared__ float sL[8];
    __shared__ float sG[8];
    const int wid = t >> 5;
    if ((t & 31) == 0) { sL[wid] = sumL; sG[wid] = sumG; }
    __syncthreads();
    float totL = 0.0f, totG = 0.0f;
#pragma unroll
    for (int w = 0; w < 8; ++w) { totL += sL[w]; totG += sG[w]; }

    const float rL = 0.5f / totL;   // (1-ALPHA)/sum_local
    const float rG = 0.5f / totG;   // ALPHA/sum_long

    float* orow = out + ((size_t)h * GA_N + i) * GA_N;
#pragma unroll
    for (int g = 0; g < 2; ++g) {
        const int j = g * 1024 + 4 * t;
        float oo[4];
#pragma unroll
        for (int q = 0; q < 4; ++q) {
            float v = 0.0f;
            if (ml[g][q] != 0.0f) v += e[g][q] * rL;
            if (mg[g][q] != 0.0f) v += e[g][q] * rG;
            oo[q] = v;
        }
        float4 o;
        o.x = oo[0]; o.y = oo[1]; o.z = oo[2]; o.w = oo[3];
        *(float4*)(orow + j) = o;
    }
}

// ---------------------------------------------------------------------------
extern "C" void kernel_launch(void* const* d_in, const int* in_sizes, int n_in,
                              void* d_out, int out_size, void* d_ws, size_t ws_size,
                              hipStream_t stream) {
    const float* localm = (const float*)d_in[0];   // [N,N]
    const float* longm  = (const float*)d_in[1];   // [N,N]
    const float* X      = (const float*)d_in[2];   // [H,N,D]
    const float* V      = (const float*)d_in[3];   // [H,2,D,1]
    float* F   = (float*)d_ws;                     // [2,H,N] = 64 KB scratch
    float* out = (float*)d_out;                    // [H,N,N]

    ga_proj_wmma<<<GA_H * (GA_N / 16), 32, 0, stream>>>(X, V, F);
    ga_att<<<GA_H * GA_N, 256, 0, stream>>>(localm, longm, F, out);
}